// GATLayer_57930518888947
// MI455X (gfx1250) — compile-verified
//
#include <hip/hip_runtime.h>
#include <hip/hip_bf16.h>
#include <cstdint>

#define F_IN       128
#define F_OUT      64
#define NEG_SLOPE  0.01f

typedef float v2f __attribute__((ext_vector_type(2)));
typedef float v8f __attribute__((ext_vector_type(8)));

// ---- order-preserving float<->uint map for atomic segment-max --------------
__device__ __forceinline__ unsigned f32_enc(float f) {
    unsigned u = __float_as_uint(f);
    return (u & 0x80000000u) ? ~u : (u | 0x80000000u);
}
__device__ __forceinline__ float f32_dec(unsigned k) {
    unsigned u = (k & 0x80000000u) ? (k & 0x7FFFFFFFu) : ~k;
    return __uint_as_float(u);
}
__device__ __forceinline__ bool finite_bits(float f) {
    return ((__float_as_uint(f) >> 23) & 0xFFu) != 0xFFu;
}

// ---- init: zero output + segment sums, set segment-max keys to minimum -----
__global__ void k_init(float* __restrict__ out, float* __restrict__ ssum,
                       unsigned* __restrict__ m_u, int n, int total) {
    int i = blockIdx.x * blockDim.x + threadIdx.x;
    if (i < total) out[i] = 0.0f;
    if (i < n) { ssum[i] = 0.0f; m_u[i] = 0u; }   // key 0 decodes to NaN -> m=0
}

// ---- fused GEMM: z = h@Wf.T and zr = h@Wfr.T via V_WMMA_F32_16X16X4_F32 ----
// Block = 128 threads = 4 waves. Block covers 16 nodes x 64 output cols;
// wave w handles output-column tile [16w, 16w+16). A-tile (h rows) is shared
// logically; each wave issues 2 WMMAs per K-step (z and zr accumulators).
__global__ __launch_bounds__(128) void k_gemm(const float* __restrict__ h,
                                              const float* __restrict__ Wf,
                                              const float* __restrict__ Wfr,
                                              float* __restrict__ z,
                                              float* __restrict__ zr,
                                              int n_nodes) {
    const int lane = threadIdx.x & 31;
    const int wave = threadIdx.x >> 5;       // output tile 0..3
    const int r16  = lane & 15;
    const int hi   = lane >> 4;              // 0: K-pair {0,1}; 1: K-pair {2,3}
    const int kh   = hi * 2;
    const long m0  = (long)blockIdx.x * 16;  // node base
    const int  n0  = wave * 16;              // output-col base

    long mrow = m0 + r16;
    if (mrow >= n_nodes) mrow = n_nodes - 1;      // clamp (no divergence: EXEC all 1s)
    const float* __restrict__ ha  = h   + mrow * F_IN;
    const float* __restrict__ wf  = Wf  + (long)(n0 + r16) * F_IN;   // B[k][n] = Wf[n][k]
    const float* __restrict__ wfr = Wfr + (long)(n0 + r16) * F_IN;

    v8f acc  = {};
    v8f accr = {};
    for (int k = 0; k < F_IN; k += 4) {
        // A 16x4 f32 layout: VGPR0 = K=kh, VGPR1 = K=kh+1 for this lane half
        v2f a  = *(const v2f*)(ha  + k + kh);
        v2f bf = *(const v2f*)(wf  + k + kh);
        v2f br = *(const v2f*)(wfr + k + kh);
        acc  = __builtin_amdgcn_wmma_f32_16x16x4_f32(false, a, false, bf,
                                                     (short)0, acc,  false, false);
        accr = __builtin_amdgcn_wmma_f32_16x16x4_f32(false, a, false, br,
                                                     (short)0, accr, false, false);
    }

    // C/D layout: VGPR r = row M=r (lanes 0-15) / M=r+8 (lanes 16-31), N = lane%16
    const int col = n0 + r16;
    for (int r = 0; r < 8; ++r) {
        long row = m0 + r + hi * 8;
        if (row < n_nodes) {
            z [row * F_OUT + col] = acc[r];
            zr[row * F_OUT + col] = accr[r];
        }
    }
}

// ---- per-node attention scalars: a1=z.Wa[:64], a2=z.Wa[64:], same for zr ---
__global__ __launch_bounds__(256) void k_node_attn(const float* __restrict__ z,
                                                   const float* __restrict__ zr,
                                                   const float* __restrict__ Wa,
                                                   float* __restrict__ a1, float* __restrict__ a2,
                                                   float* __restrict__ ar1, float* __restrict__ ar2,
                                                   int n_nodes) {
    int i = blockIdx.x * blockDim.x + threadIdx.x;
    if (i >= n_nodes) return;
    const float* __restrict__ zrow  = z  + (long)i * F_OUT;
    const float* __restrict__ zrrow = zr + (long)i * F_OUT;
    float s1 = 0.f, s2 = 0.f, sr1 = 0.f, sr2 = 0.f;
#pragma unroll 8
    for (int j = 0; j < F_OUT; ++j) {
        float wa = Wa[j], wb = Wa[F_OUT + j];
        float zv = zrow[j], zrv = zrrow[j];
        s1  = fmaf(zv,  wa, s1);
        s2  = fmaf(zv,  wb, s2);
        sr1 = fmaf(zrv, wa, sr1);
        sr2 = fmaf(zrv, wb, sr2);
    }
    a1[i] = s1; a2[i] = s2; ar1[i] = sr1; ar2[i] = sr2;
}

// ---- edge pass 1: e_msg + atomic segment max over dst ----------------------
__global__ __launch_bounds__(256) void k_edge_emsg(const float* __restrict__ a1,
                                                   const float* __restrict__ a2,
                                                   const float* __restrict__ ar1,
                                                   const float* __restrict__ ar2,
                                                   const float* __restrict__ dir,
                                                   const int* __restrict__ src,
                                                   const int* __restrict__ dst,
                                                   float* __restrict__ e_msg,
                                                   unsigned* __restrict__ m_u,
                                                   int n_edges) {
    int k = blockIdx.x * blockDim.x + threadIdx.x;
    if (k >= n_edges) return;
    int sn = src[k], dn = dst[k];
    float d0 = dir[2 * k], d1 = dir[2 * k + 1];
    float e  = a1[sn]  + a2[dn];
    e  = (e  > 0.f) ? e  : NEG_SLOPE * e;
    float er = ar1[sn] + ar2[dn];
    er = (er > 0.f) ? er : NEG_SLOPE * er;
    float em = d0 * e + d1 * er;
    e_msg[k] = em;
    atomicMax(&m_u[dn], f32_enc(em));
}

// ---- decode segment max; empty / non-finite segments -> 0 ------------------
__global__ void k_decode_max(const unsigned* __restrict__ m_u,
                             float* __restrict__ m, int n_nodes) {
    int i = blockIdx.x * blockDim.x + threadIdx.x;
    if (i >= n_nodes) return;
    float mf = f32_dec(m_u[i]);
    m[i] = finite_bits(mf) ? mf : 0.0f;
}

// ---- edge pass 2: w = exp(e_msg - m[dst]); segment sum ---------------------
__global__ __launch_bounds__(256) void k_edge_exp(const float* __restrict__ e_msg,
                                                  const float* __restrict__ m,
                                                  const int* __restrict__ dst,
                                                  float* __restrict__ w,
                                                  float* __restrict__ ssum,
                                                  int n_edges) {
    int k = blockIdx.x * blockDim.x + threadIdx.x;
    if (k >= n_edges) return;
    int dn = dst[k];
    float wk = expf(e_msg[k] - m[dn]);
    w[k] = wk;
    atomicAdd(&ssum[dn], wk);
}

// ---- edge pass 3: out[dst] += alpha * (d0*z[src] + d1*zr[src]) -------------
// 64 threads per edge (thread j handles feature j); 4 edges per 256-block.
__global__ __launch_bounds__(256) void k_edge_aggr(const float* __restrict__ z,
                                                   const float* __restrict__ zr,
                                                   const float* __restrict__ dir,
                                                   const int* __restrict__ src,
                                                   const int* __restrict__ dst,
                                                   const float* __restrict__ w,
                                                   const float* __restrict__ ssum,
                                                   float* __restrict__ out,
                                                   int n_edges) {
    long e = (long)blockIdx.x * 4 + (threadIdx.x >> 6);
    int  j = threadIdx.x & 63;
    if (e >= n_edges) return;
    int sn = src[e], dn = dst[e];
    float d0 = dir[2 * e], d1 = dir[2 * e + 1];
    float alpha = w[e] / fmaxf(ssum[dn], 1e-9f);
    float q = d0 * z[(long)sn * F_OUT + j] + d1 * zr[(long)sn * F_OUT + j];
    atomicAdd(out + (long)dn * F_OUT + j, alpha * q);
}

extern "C" void kernel_launch(void* const* d_in, const int* in_sizes, int n_in,
                              void* d_out, int out_size, void* d_ws, size_t ws_size,
                              hipStream_t stream) {
    const float* h   = (const float*)d_in[0];   // (N,128)
    const float* Wf  = (const float*)d_in[1];   // (64,128)
    const float* Wfr = (const float*)d_in[2];   // (64,128)
    const float* Wa  = (const float*)d_in[3];   // (1,128)
    const float* dir = (const float*)d_in[4];   // (E,2,1)
    const int*   src = (const int*)d_in[5];     // (E,)
    const int*   dst = (const int*)d_in[6];     // (E,)

    const int N = in_sizes[0] / F_IN;
    const int E = in_sizes[5];

    // workspace carve-up (floats / uints)
    float*    z     = (float*)d_ws;
    float*    zr    = z   + (size_t)N * F_OUT;
    float*    a1    = zr  + (size_t)N * F_OUT;
    float*    a2    = a1  + N;
    float*    ar1   = a2  + N;
    float*    ar2   = ar1 + N;
    unsigned* m_u   = (unsigned*)(ar2 + N);
    float*    m     = (float*)(m_u + N);
    float*    ssum  = m + N;
    float*    e_msg = ssum + N;
    float*    w     = e_msg + E;

    float* out = (float*)d_out;
    const int total = N * F_OUT;

    k_init<<<(total + 255) / 256, 256, 0, stream>>>(out, ssum, m_u, N, total);

    k_gemm<<<(N + 15) / 16, 128, 0, stream>>>(h, Wf, Wfr, z, zr, N);

    k_node_attn<<<(N + 255) / 256, 256, 0, stream>>>(z, zr, Wa, a1, a2, ar1, ar2, N);

    k_edge_emsg<<<(E + 255) / 256, 256, 0, stream>>>(a1, a2, ar1, ar2, dir, src, dst,
                                                     e_msg, m_u, E);

    k_decode_max<<<(N + 255) / 256, 256, 0, stream>>>(m_u, m, N);

    k_edge_exp<<<(E + 255) / 256, 256, 0, stream>>>(e_msg, m, dst, w, ssum, E);

    k_edge_aggr<<<((E + 3) / 4), 256, 0, stream>>>(z, zr, dir, src, dst, w, ssum, out, E);
}